// StrokeLoss_63505386439206
// MI455X (gfx1250) — compile-verified
//
#include <hip/hip_runtime.h>
#include <math.h>

// Problem constants (match reference)
#define B_    8
#define D_    16
#define H_    512
#define W_    512
#define HW    (H_*W_)
#define NSEG  33

// Workspace layout (floats)
#define OFF_CNT    0                       // [B][NSEG]           264
#define OFF_SUMS   264                     // [B][NSEG][D]        4224
#define OFF_HINGE  4488                    // [B][NSEG]           264
#define OFF_MU     4752                    // [B][NSEG][D]        4224
#define OFF_SC     8976                    // scalars             32
#define WS_FLOATS  9008
// scalar slots: 0 bce, 1 focal_end, 2 focal_junc, 3 ori_num, 4 ori_den,
//               5..12 inter[b], 13..20 psum[b], 21..28 tsum[b]

typedef __attribute__((ext_vector_type(2))) float v2f;
typedef __attribute__((ext_vector_type(8))) float v8f;

__device__ __forceinline__ void ld4(const float* __restrict__ p, float v[4]) {
  float4 t = *reinterpret_cast<const float4*>(p);
  v[0] = t.x; v[1] = t.y; v[2] = t.z; v[3] = t.w;
}

__device__ __forceinline__ float wave_reduce(float v) {
#pragma unroll
  for (int off = 16; off > 0; off >>= 1) v += __shfl_down(v, off, 32);
  return v;
}

__global__ void zero_ws_kernel(float* __restrict__ ws) {
  int i = blockIdx.x * blockDim.x + threadIdx.x;
  if (i < WS_FLOATS) ws[i] = 0.f;
}

// ---------------------------------------------------------------------------
// Pass 1: all elementwise loss reductions + per-segment count/sum of embedding
// grid (HW/1024, B), block 256, 4 pixels/thread (float4 loads)
// ---------------------------------------------------------------------------
__global__ void pass1_kernel(const float* __restrict__ fg_logit,
                             const float* __restrict__ emb,
                             const float* __restrict__ pred_ori,
                             const float* __restrict__ end_logit,
                             const float* __restrict__ junc_logit,
                             const float* __restrict__ fg_mask,
                             const int*  __restrict__ ids,
                             const float* __restrict__ tgt_ori,
                             const float* __restrict__ end_heat,
                             const float* __restrict__ junc_heat,
                             float* __restrict__ ws) {
  __shared__ float cnt_s[NSEG];
  __shared__ float sums_s[NSEG * D_];
  const int tid = threadIdx.x;
  const int b = blockIdx.y;
  for (int i = tid; i < NSEG * (D_ + 1); i += blockDim.x) {
    if (i < NSEG) cnt_s[i] = 0.f; else sums_s[i - NSEG] = 0.f;
  }
  __syncthreads();

  const int p0 = (blockIdx.x * blockDim.x + tid) * 4;
  const long base = (long)b * HW + p0;

  float xfg[4], tfg[4], xe[4], te[4], xj[4], tjh[4];
  float po0[4], po1[4], go0[4], go1[4];
  ld4(fg_logit + base, xfg);
  ld4(fg_mask + base, tfg);
  ld4(end_logit + base, xe);
  ld4(end_heat + base, te);
  ld4(junc_logit + base, xj);
  ld4(junc_heat + base, tjh);
  ld4(pred_ori + ((long)b * 2 + 0) * HW + p0, po0);
  ld4(pred_ori + ((long)b * 2 + 1) * HW + p0, po1);
  ld4(tgt_ori  + ((long)b * 2 + 0) * HW + p0, go0);
  ld4(tgt_ori  + ((long)b * 2 + 1) * HW + p0, go1);
  int4 idv = *reinterpret_cast<const int4*>(ids + base);
  int id4[4] = { idv.x, idv.y, idv.z, idv.w };

  float bce_acc = 0.f, inter_acc = 0.f, p_acc = 0.f, t_acc = 0.f;
  float fe_acc = 0.f, fj_acc = 0.f, on_acc = 0.f, od_acc = 0.f;
  float w4[4];
#pragma unroll
  for (int j = 0; j < 4; ++j) {
    float t = tfg[j];
    float x = xfg[j];
    // fg BCE + dice partials
    float bce = fmaxf(x, 0.f) - x * t + log1pf(expf(-fabsf(x)));
    bce_acc += bce;
    float p = 1.f / (1.f + expf(-x));
    inter_acc += p * t; p_acc += p; t_acc += t;
    // focal (endpoint)
    {
      float tt = fminf(fmaxf(te[j], 0.f), 1.f);
      float xx = xe[j];
      float bb = fmaxf(xx, 0.f) - xx * tt + log1pf(expf(-fabsf(xx)));
      float pt = expf(-bb);
      float at = 0.25f * tt + 0.75f * (1.f - tt);
      float om = 1.f - pt;
      fe_acc += at * om * om * bb;
    }
    // focal (junction)
    {
      float tt = fminf(fmaxf(tjh[j], 0.f), 1.f);
      float xx = xj[j];
      float bb = fmaxf(xx, 0.f) - xx * tt + log1pf(expf(-fabsf(xx)));
      float pt = expf(-bb);
      float at = 0.25f * tt + 0.75f * (1.f - tt);
      float om = 1.f - pt;
      fj_acc += at * om * om * bb;
    }
    // orientation
    float m = (t > 0.5f) ? 1.f : 0.f;
    {
      float pn = fmaxf(sqrtf(po0[j] * po0[j] + po1[j] * po1[j]), 1e-8f);
      float gn = fmaxf(sqrtf(go0[j] * go0[j] + go1[j] * go1[j]), 1e-8f);
      float dt = (po0[j] * go0[j] + po1[j] * go1[j]) / (pn * gn);
      on_acc += (1.f - fabsf(dt)) * m;
      od_acc += m;
    }
    // discriminative validity
    float valid = (m > 0.f && id4[j] > 0) ? 1.f : 0.f;
    w4[j] = valid;
    if (valid > 0.f) atomicAdd(&cnt_s[id4[j]], 1.f);
  }

  // segment sums of embedding (stream channels, coalesced b128 per channel)
#pragma unroll
  for (int d = 0; d < D_; ++d) {
    float ev[4];
    ld4(emb + ((long)b * D_ + d) * HW + p0, ev);
#pragma unroll
    for (int j = 0; j < 4; ++j)
      if (w4[j] > 0.f) atomicAdd(&sums_s[id4[j] * D_ + d], ev[j]);
  }

  // wave-level scalar reduction, one global atomic per wave per scalar
  float* sc = ws + OFF_SC;
  bce_acc = wave_reduce(bce_acc);
  inter_acc = wave_reduce(inter_acc);
  p_acc = wave_reduce(p_acc);
  t_acc = wave_reduce(t_acc);
  fe_acc = wave_reduce(fe_acc);
  fj_acc = wave_reduce(fj_acc);
  on_acc = wave_reduce(on_acc);
  od_acc = wave_reduce(od_acc);
  if ((tid & 31) == 0) {
    atomicAdd(&sc[0], bce_acc);
    atomicAdd(&sc[1], fe_acc);
    atomicAdd(&sc[2], fj_acc);
    atomicAdd(&sc[3], on_acc);
    atomicAdd(&sc[4], od_acc);
    atomicAdd(&sc[5 + b], inter_acc);
    atomicAdd(&sc[13 + b], p_acc);
    atomicAdd(&sc[21 + b], t_acc);
  }
  __syncthreads();
  for (int i = tid; i < NSEG; i += blockDim.x)
    atomicAdd(&ws[OFF_CNT + b * NSEG + i], cnt_s[i]);
  for (int i = tid; i < NSEG * D_; i += blockDim.x)
    atomicAdd(&ws[OFF_SUMS + b * NSEG * D_ + i], sums_s[i]);
}

// ---------------------------------------------------------------------------
// mu = sums / max(cnt,1)
// ---------------------------------------------------------------------------
__global__ void mu_kernel(float* __restrict__ ws) {
  int i = blockIdx.x * blockDim.x + threadIdx.x;
  if (i < B_ * NSEG) {
    float inv = 1.f / fmaxf(ws[OFF_CNT + i], 1.f);
#pragma unroll
    for (int d = 0; d < D_; ++d)
      ws[OFF_MU + i * D_ + d] = ws[OFF_SUMS + i * D_ + d] * inv;
  }
}

// ---------------------------------------------------------------------------
// Pass 2: hinge = max(||mu[seg]-e|| - 0.5, 0)^2 summed per segment
// ---------------------------------------------------------------------------
__global__ void pass2_kernel(const float* __restrict__ emb,
                             const float* __restrict__ fg_mask,
                             const int*  __restrict__ ids,
                             float* __restrict__ ws) {
  __shared__ float mu_s[NSEG * D_];
  __shared__ float hinge_s[NSEG];
  const int tid = threadIdx.x;
  const int b = blockIdx.y;
  for (int i = tid; i < NSEG * D_; i += blockDim.x) mu_s[i] = ws[OFF_MU + b * NSEG * D_ + i];
  for (int i = tid; i < NSEG; i += blockDim.x) hinge_s[i] = 0.f;
  __syncthreads();

  const int p0 = (blockIdx.x * blockDim.x + tid) * 4;
  const long base = (long)b * HW + p0;
  float tfg[4];
  ld4(fg_mask + base, tfg);
  int4 idv = *reinterpret_cast<const int4*>(ids + base);
  int id4[4] = { idv.x, idv.y, idv.z, idv.w };
  float d2[4] = { 0.f, 0.f, 0.f, 0.f };
#pragma unroll
  for (int d = 0; d < D_; ++d) {
    float ev[4];
    ld4(emb + ((long)b * D_ + d) * HW + p0, ev);
#pragma unroll
    for (int j = 0; j < 4; ++j) {
      float df = mu_s[id4[j] * D_ + d] - ev[j];
      d2[j] += df * df;
    }
  }
#pragma unroll
  for (int j = 0; j < 4; ++j) {
    if (tfg[j] > 0.5f && id4[j] > 0) {
      float dist = sqrtf(fmaxf(d2[j], 1e-12f));
      float h = fmaxf(dist - 0.5f, 0.f);
      atomicAdd(&hinge_s[id4[j]], h * h);
    }
  }
  __syncthreads();
  for (int i = tid; i < NSEG; i += blockDim.x)
    atomicAdd(&ws[OFF_HINGE + b * NSEG + i], hinge_s[i]);
}

// ---------------------------------------------------------------------------
// Final: Gram matrix of cluster means via V_WMMA_F32_16X16X4_F32 (one wave),
// pairwise-distance hinge, L_var/L_reg, and assembly of the total loss.
// Mu padded to 48x16 -> 3 row tiles. G = M*M^T so A and B per-lane data are
// identical (lane m holds row m, K split across lane halves per ISA layout).
// ---------------------------------------------------------------------------
__global__ void final_kernel(float* __restrict__ ws, float* __restrict__ out) {
  __shared__ float mu_s[NSEG * D_];
  __shared__ float norm2_s[48];
  __shared__ float vk_s[NSEG];
  const int lane = threadIdx.x;
  const int half = lane >> 4;     // 0: K+0/1, 1: K+2/3 in A layout
  const int lm = lane & 15;

  float acc_loss = 0.f, acc_has = 0.f;

  for (int b = 0; b < B_; ++b) {
    __syncthreads();
    for (int i = lane; i < NSEG * D_; i += 32) mu_s[i] = ws[OFF_MU + b * NSEG * D_ + i];
    for (int i = lane; i < NSEG; i += 32)
      vk_s[i] = (i > 0 && ws[OFF_CNT + b * NSEG + i] > 0.f) ? 1.f : 0.f;
    for (int i = lane; i < 48; i += 32) norm2_s[i] = 0.f;
    __syncthreads();

    // Preload A operands: 3 row-tiles x 4 K-chunks (16x4 f32 A layout)
    v2f A[3][4];
#pragma unroll
    for (int t = 0; t < 3; ++t) {
      int row = t * 16 + lm;
#pragma unroll
      for (int c = 0; c < 4; ++c) {
        int k = c * 4 + half * 2;
        float v0 = 0.f, v1 = 0.f;
        if (row < NSEG) { v0 = mu_s[row * D_ + k]; v1 = mu_s[row * D_ + k + 1]; }
        v2f a = { v0, v1 };
        A[t][c] = a;
      }
    }

    // Diagonal Gram tiles -> squared norms
    v8f gd[3];
#pragma unroll
    for (int t = 0; t < 3; ++t) {
      v8f acc = {};
#pragma unroll
      for (int c = 0; c < 4; ++c)
        acc = __builtin_amdgcn_wmma_f32_16x16x4_f32(
            false, A[t][c], false, A[t][c], (short)0, acc, false, false);
      gd[t] = acc;
#pragma unroll
      for (int r = 0; r < 8; ++r) {
        int M = r + 8 * half;           // C/D layout: reg r, lane half -> row
        if (M == lm) norm2_s[t * 16 + M] = acc[r];
      }
    }
    __syncthreads();

    // Pairwise hinge: sum over i<j valid of max(3 - ||mu_i-mu_j||, 0)^2
    float psum = 0.f;
#pragma unroll
    for (int ti = 0; ti < 3; ++ti) {
#pragma unroll
      for (int tj = ti; tj < 3; ++tj) {
        v8f g;
        if (ti == tj) {
          g = gd[ti];
        } else {
          v8f acc = {};
#pragma unroll
          for (int c = 0; c < 4; ++c)
            acc = __builtin_amdgcn_wmma_f32_16x16x4_f32(
                false, A[ti][c], false, A[tj][c], (short)0, acc, false, false);
          g = acc;
        }
#pragma unroll
        for (int r = 0; r < 8; ++r) {
          int i = ti * 16 + r + 8 * half;
          int j = tj * 16 + lm;
          if (i < j && j < NSEG && vk_s[i] > 0.f && vk_s[j] > 0.f) {
            float dd2 = norm2_s[i] + norm2_s[j] - 2.f * g[r];
            float dp = sqrtf(fmaxf(dd2, 1e-12f));
            float h = fmaxf(3.0f - dp, 0.f);   // 2*DELTA_D = 3.0
            psum += h * h;
          }
        }
      }
    }
    psum = wave_reduce(psum);

    if (lane == 0) {
      float K = 0.f, Lvar = 0.f, Lreg = 0.f;
      for (int k = 1; k < NSEG; ++k) {
        float c = ws[OFF_CNT + b * NSEG + k];
        if (c > 0.f) {
          K += 1.f;
          Lvar += ws[OFF_HINGE + b * NSEG + k] / fmaxf(c, 1.f);
          Lreg += sqrtf(fmaxf(norm2_s[k], 1e-12f));
        }
      }
      float Kc = fmaxf(K, 1.f);
      Lvar /= Kc;
      Lreg /= Kc;
      float Ldist = psum / fmaxf(K * (K - 1.f) * 0.5f, 1.f);
      float has = (K > 0.f) ? 1.f : 0.f;
      acc_loss += (Lvar + Ldist + 0.001f * Lreg) * has;
      acc_has += has;
    }
  }

  if (lane == 0) {
    float* sc = ws + OFF_SC;
    const float inv = 1.f / (float)(B_ * HW);
    float bce_mean = sc[0] * inv;
    float dice = 0.f;
    for (int b = 0; b < B_; ++b)
      dice += 1.f - (2.f * sc[5 + b] + 1.f) / (sc[13 + b] + sc[21 + b] + 1.f);
    dice *= (1.f / (float)B_);
    float L_fg = bce_mean + dice;
    float L_end = sc[1] * inv;
    float L_junc = sc[2] * inv;
    float L_ori = sc[3] / fmaxf(sc[4], 1.f);
    float L_emb = (acc_has > 0.f) ? acc_loss / fmaxf(acc_has, 1.f) : 0.f;
    out[0] = 1.0f * L_fg + 1.0f * L_emb + 0.5f * L_ori + 0.5f * L_end + 0.5f * L_junc;
  }
}

// ---------------------------------------------------------------------------
extern "C" void kernel_launch(void* const* d_in, const int* in_sizes, int n_in,
                              void* d_out, int out_size, void* d_ws, size_t ws_size,
                              hipStream_t stream) {
  const float* fg_logit   = (const float*)d_in[0];
  const float* emb        = (const float*)d_in[1];
  const float* pred_ori   = (const float*)d_in[2];
  const float* end_logit  = (const float*)d_in[3];
  const float* junc_logit = (const float*)d_in[4];
  const float* fg_mask    = (const float*)d_in[5];
  const int*   ids        = (const int*)d_in[6];
  const float* tgt_ori    = (const float*)d_in[7];
  const float* end_heat   = (const float*)d_in[8];
  const float* junc_heat  = (const float*)d_in[9];
  float* ws  = (float*)d_ws;
  float* out = (float*)d_out;

  zero_ws_kernel<<<(WS_FLOATS + 255) / 256, 256, 0, stream>>>(ws);

  dim3 grid(HW / (256 * 4), B_);   // 256 blocks x 8 batches, 4 px/thread
  pass1_kernel<<<grid, 256, 0, stream>>>(fg_logit, emb, pred_ori, end_logit,
                                         junc_logit, fg_mask, ids, tgt_ori,
                                         end_heat, junc_heat, ws);
  mu_kernel<<<(B_ * NSEG + 255) / 256, 256, 0, stream>>>(ws);
  pass2_kernel<<<grid, 256, 0, stream>>>(emb, fg_mask, ids, ws);
  final_kernel<<<1, 32, 0, stream>>>(ws, out);
}